// HGTVulNodeClassifier_60550448939646
// MI455X (gfx1250) — compile-verified
//
#include <hip/hip_runtime.h>
#include <math.h>

// Problem constants (from reference)
#define T_  2
#define N_  50000
#define D_  256
#define H_  8
#define DK_ 32
#define R_  4
#define L_  2
#define E_  500000

#define ND_  (N_ * D_)          // 12,800,000
#define TND_ (T_ * N_ * D_)     // 25,600,000
#define NH_  (N_ * H_)          // 400,000
#define EH_  (E_ * H_)          // 4,000,000

typedef float v2f __attribute__((ext_vector_type(2)));
typedef float v8f __attribute__((ext_vector_type(8)));

// ---------- helpers ----------
__device__ __forceinline__ unsigned f32_order_enc(float f) {
    unsigned u = __float_as_uint(f);
    return (u & 0x80000000u) ? ~u : (u | 0x80000000u);
}
__device__ __forceinline__ float f32_order_dec(unsigned u) {
    return __uint_as_float((u & 0x80000000u) ? (u & 0x7fffffffu) : ~u);
}
__device__ __forceinline__ void atomic_add_f32(float* p, float v) {
    unsafeAtomicAdd(p, v);   // lowers to global_atomic_add_f32 on gfx1250
}

// CDNA5 async global->LDS copy (ASYNCcnt-tracked, no VGPR transit).
// LDS operand is the wave-relative LDS byte address = low 32 bits of the
// generic shared pointer ({SHARED_BASE, offset} aperture encoding).
__device__ __forceinline__ void async_g2l_b128(void* lds, const void* gptr) {
    asm volatile("global_load_async_to_lds_b128 %0, %1, off"
                 :
                 : "v"((unsigned)(unsigned long long)(uintptr_t)lds),
                   "v"((unsigned long long)(uintptr_t)gptr)
                 : "memory");
}
__device__ __forceinline__ void wait_async_all() {
    asm volatile("s_wait_asynccnt 0" ::: "memory");
}

// ---------- fill kernels ----------
__global__ void fill_f32_kernel(float* __restrict__ p, float v, int n) {
    int i = blockIdx.x * 256 + threadIdx.x;
    if (i < n) p[i] = v;
}
__global__ void fill_u32_kernel(unsigned* __restrict__ p, unsigned v, int n) {
    int i = blockIdx.x * 256 + threadIdx.x;
    if (i < n) p[i] = v;
}

// ---------- adapt: h = gelu(feat @ adapt_W + adapt_b) ----------
__global__ void adapt_kernel(const float* __restrict__ feat,      // [T,N,T]
                             const float* __restrict__ aW,        // [T,T,D]
                             const float* __restrict__ ab,        // [T,D]
                             float* __restrict__ h) {             // [T,N,D]
    int idx = blockIdx.x * 256 + threadIdx.x;
    if (idx >= TND_) return;
    int d  = idx & (D_ - 1);
    int tn = idx >> 8;           // t*N + n
    int t  = tn / N_;
    float f0 = feat[(size_t)tn * T_ + 0];
    float f1 = feat[(size_t)tn * T_ + 1];
    float x = f0 * aW[(t * T_ + 0) * D_ + d] + f1 * aW[(t * T_ + 1) * D_ + d]
            + ab[t * D_ + d];
    h[idx] = 0.5f * x * (1.0f + erff(x * 0.70710678118654752f)); // exact gelu
}

// ---------- dense GEMM: Y[nrows,256] = xscale*(X @ W) + bias ----------
// block = 256 threads = 8 waves arranged 2(M)x4(N); block tile 32x64.
// X/W chunks staged via GLOBAL_LOAD_ASYNC_TO_LDS_B128 (ASYNCcnt), frags read
// back with ds loads, math on V_WMMA_F32_16X16X4_F32.
__global__ __launch_bounds__(256) void gemm_bias_wmma(
    const float* __restrict__ X, const float* __restrict__ W,
    const float* __restrict__ bias, float* __restrict__ Y,
    int nrows, float xscale) {
    __shared__ float xs[32][68];
    __shared__ float ws[64][68];
    const int tid   = threadIdx.x;
    const int mBase = blockIdx.x * 32;
    const int nBase = blockIdx.y * 64;
    const int wave  = tid >> 5, lane = tid & 31;
    const int wm    = wave >> 2, wn = wave & 3;
    const int lrow  = lane & 15;
    const bool hi   = lane >= 16;

    v8f acc = {};
    for (int kk = 0; kk < D_; kk += 64) {
        // stage X chunk [32 x 64] (clamped rows for the ragged tail tile)
#pragma unroll
        for (int i = 0; i < 2; ++i) {
            int idx = tid + i * 256;          // float4 index
            int row = idx >> 4;
            int c4  = (idx & 15) << 2;
            int gr  = mBase + row; if (gr > nrows - 1) gr = nrows - 1;
            async_g2l_b128(&xs[row][c4], X + (size_t)gr * D_ + kk + c4);
        }
        // stage W chunk [64 x 64]
#pragma unroll
        for (int i = 0; i < 4; ++i) {
            int idx = tid + i * 256;
            int row = idx >> 4;
            int c4  = (idx & 15) << 2;
            async_g2l_b128(&ws[row][c4], W + (size_t)(kk + row) * D_ + nBase + c4);
        }
        wait_async_all();
        __syncthreads();
#pragma unroll
        for (int ks = 0; ks < 64; ks += 4) {
            const int ka = ks + (hi ? 2 : 0);
            v2f a, b;
            a.x = xs[wm * 16 + lrow][ka];     a.y = xs[wm * 16 + lrow][ka + 1];
            b.x = ws[ka][wn * 16 + lrow];     b.y = ws[ka + 1][wn * 16 + lrow];
            acc = __builtin_amdgcn_wmma_f32_16x16x4_f32(
                false, a, false, b, (short)0, acc, false, false);
        }
        __syncthreads();
    }
    const int col   = nBase + wn * 16 + lrow;
    const float bv  = bias[col];
    const int rbase = mBase + wm * 16 + (hi ? 8 : 0);
#pragma unroll
    for (int i = 0; i < 8; ++i) {
        int r = rbase + i;
        if (r < nrows) Y[(size_t)r * D_ + col] = acc[i] * xscale + bv;
    }
}

// ---------- per-head relation transform: Y[n, h*32+j] = sum_i X[n,h*32+i] * Wr[h,i,j] ----------
// grid = (ceil(N/32), H); block 128 threads = 4 waves arranged 2(M)x2(N); K=32.
__global__ __launch_bounds__(128) void rel_gemm_wmma(
    const float* __restrict__ X,   // [N, D] (head-blocked)
    const float* __restrict__ Wr,  // [H, 32, 32]
    float* __restrict__ Y,         // [N, D]
    int nrows) {
    __shared__ float xs[32][36];
    __shared__ float ws[32][36];
    const int tid   = threadIdx.x;
    const int hh    = blockIdx.y;
    const int mBase = blockIdx.x * 32;
    const int wave  = tid >> 5, lane = tid & 31;
    const int wm    = wave >> 1, wn = wave & 1;
    const int lrow  = lane & 15;
    const bool hi   = lane >= 16;

#pragma unroll
    for (int i = 0; i < 2; ++i) {
        int idx = tid + i * 128;   // float4 index, 0..255
        int row = idx >> 3;
        int c4  = (idx & 7) << 2;
        int gr  = mBase + row; if (gr > nrows - 1) gr = nrows - 1;
        async_g2l_b128(&xs[row][c4], X + (size_t)gr * D_ + hh * DK_ + c4);
    }
#pragma unroll
    for (int i = 0; i < 2; ++i) {
        int idx = tid + i * 128;
        int row = idx >> 3;
        int c4  = (idx & 7) << 2;
        async_g2l_b128(&ws[row][c4], Wr + hh * DK_ * DK_ + row * DK_ + c4);
    }
    wait_async_all();
    __syncthreads();
    v8f acc = {};
#pragma unroll
    for (int ks = 0; ks < DK_; ks += 4) {
        const int ka = ks + (hi ? 2 : 0);
        v2f a, b;
        a.x = xs[wm * 16 + lrow][ka];  a.y = xs[wm * 16 + lrow][ka + 1];
        b.x = ws[ka][wn * 16 + lrow];  b.y = ws[ka + 1][wn * 16 + lrow];
        acc = __builtin_amdgcn_wmma_f32_16x16x4_f32(
            false, a, false, b, (short)0, acc, false, false);
    }
    const int col   = hh * DK_ + wn * 16 + lrow;
    const int rbase = mBase + wm * 16 + (hi ? 8 : 0);
#pragma unroll
    for (int i = 0; i < 8; ++i) {
        int r = rbase + i;
        if (r < nrows) Y[(size_t)r * D_ + col] = acc[i];
    }
}

// ---------- edge score: one wave per edge (lane covers 8 channels; head = lane/4) ----------
__global__ __launch_bounds__(256) void score_kernel(
    const float* __restrict__ q,   // [N,D] dst-type
    const float* __restrict__ kr,  // [N,D]
    const int* __restrict__ si, const int* __restrict__ di,
    const float* __restrict__ pri, // [H]
    float* __restrict__ score,     // [E,H]
    unsigned* __restrict__ mbuf,   // [N,H] order-encoded running max
    int nedges) {
    int e = blockIdx.x * 8 + (threadIdx.x >> 5);
    if (e >= nedges) return;
    int lane = threadIdx.x & 31;
    int s = si[e], d = di[e];
    const float4* qp = (const float4*)(q  + (size_t)d * D_ + lane * 8);
    const float4* kp = (const float4*)(kr + (size_t)s * D_ + lane * 8);
    float4 q0 = qp[0], q1 = qp[1], k0 = kp[0], k1 = kp[1];
    float sum = q0.x * k0.x + q0.y * k0.y + q0.z * k0.z + q0.w * k0.w
              + q1.x * k1.x + q1.y * k1.y + q1.z * k1.z + q1.w * k1.w;
    sum += __shfl_xor(sum, 1, 32);
    sum += __shfl_xor(sum, 2, 32);
    int hh = lane >> 2;
    if ((lane & 3) == 0) {
        float sc = sum * pri[hh] * 0.17677669529663687f; // 1/sqrt(32)
        score[(size_t)e * H_ + hh] = sc;
        atomicMax(mbuf + (size_t)d * H_ + hh, f32_order_enc(sc));
    }
}

// ---------- ex = exp(score - m[di]); den[di] += ex ----------
__global__ __launch_bounds__(256) void exp_den_kernel(
    float* __restrict__ score,            // [E,H] in: score, out: ex
    const unsigned* __restrict__ mbuf,    // [N,H]
    float* __restrict__ den,              // [N,H]
    const int* __restrict__ di, int nedges) {
    int idx = blockIdx.x * 256 + threadIdx.x;
    if (idx >= nedges * H_) return;
    int e = idx >> 3, hh = idx & 7;
    int d = di[e];
    float m  = f32_order_dec(mbuf[(size_t)d * H_ + hh]);
    float ex = __expf(score[idx] - m);
    score[idx] = ex;
    atomic_add_f32(den + (size_t)d * H_ + hh, ex);
}

// ---------- agg[di] += (ex/den[di]) * vr[si] : one wave per edge ----------
__global__ __launch_bounds__(256) void aggregate_kernel(
    const float* __restrict__ ex,   // [E,H]
    const float* __restrict__ den,  // [N,H]
    const float* __restrict__ vr,   // [N,D]
    const int* __restrict__ si, const int* __restrict__ di,
    float* __restrict__ agg,        // [N,D] dst-type slice
    int nedges) {
    int e = blockIdx.x * 8 + (threadIdx.x >> 5);
    if (e >= nedges) return;
    int lane = threadIdx.x & 31;
    int hh = lane >> 2;
    int s = si[e], d = di[e];
    float w = ex[(size_t)e * H_ + hh] / den[(size_t)d * H_ + hh];
    const float4* vp = (const float4*)(vr + (size_t)s * D_ + lane * 8);
    float4 v0 = vp[0], v1 = vp[1];
    float* dst = agg + (size_t)d * D_ + lane * 8;
    atomic_add_f32(dst + 0, w * v0.x); atomic_add_f32(dst + 1, w * v0.y);
    atomic_add_f32(dst + 2, w * v0.z); atomic_add_f32(dst + 3, w * v0.w);
    atomic_add_f32(dst + 4, w * v1.x); atomic_add_f32(dst + 5, w * v1.y);
    atomic_add_f32(dst + 6, w * v1.z); atomic_add_f32(dst + 7, w * v1.w);
}

// ---------- skip-mix + LayerNorm : one wave per node ----------
__global__ __launch_bounds__(256) void skip_ln_kernel(
    const float* __restrict__ trans,  // [T,N,D]
    float* __restrict__ h,            // [T,N,D] (read old, write new)
    const float* __restrict__ skipv,  // [T] (layer slice)
    const float* __restrict__ lng,    // [T,D]
    const float* __restrict__ lnb) {  // [T,D]
    int node = blockIdx.x * 8 + (threadIdx.x >> 5);
    if (node >= T_ * N_) return;
    int lane = threadIdx.x & 31;
    int t = node / N_;
    float alpha = 1.0f / (1.0f + __expf(-skipv[t]));
    const float4* tp = (const float4*)(trans + (size_t)node * D_ + lane * 8);
    const float4* hp = (const float4*)(h     + (size_t)node * D_ + lane * 8);
    float4 t0 = tp[0], t1 = tp[1], h0 = hp[0], h1 = hp[1];
    float o[8];
    o[0] = t0.x * alpha + h0.x * (1.f - alpha);
    o[1] = t0.y * alpha + h0.y * (1.f - alpha);
    o[2] = t0.z * alpha + h0.z * (1.f - alpha);
    o[3] = t0.w * alpha + h0.w * (1.f - alpha);
    o[4] = t1.x * alpha + h1.x * (1.f - alpha);
    o[5] = t1.y * alpha + h1.y * (1.f - alpha);
    o[6] = t1.z * alpha + h1.z * (1.f - alpha);
    o[7] = t1.w * alpha + h1.w * (1.f - alpha);
    float s = 0.f, s2 = 0.f;
#pragma unroll
    for (int i = 0; i < 8; ++i) { s += o[i]; s2 += o[i] * o[i]; }
#pragma unroll
    for (int off = 16; off; off >>= 1) {
        s  += __shfl_xor(s,  off, 32);
        s2 += __shfl_xor(s2, off, 32);
    }
    float mean = s * (1.0f / D_);
    float var  = s2 * (1.0f / D_) - mean * mean;
    float rstd = rsqrtf(var + 1e-5f);
    const float* g = lng + t * D_ + lane * 8;
    const float* b = lnb + t * D_ + lane * 8;
    float* outp = h + (size_t)node * D_ + lane * 8;
#pragma unroll
    for (int i = 0; i < 8; ++i)
        outp[i] = (o[i] - mean) * rstd * g[i] + b[i];
}

// ---------- classifier: logits[n, 0..1] = h[n,:] @ cls_W + cls_b ----------
__global__ __launch_bounds__(256) void classifier_kernel(
    const float* __restrict__ h, const float* __restrict__ cw,
    const float* __restrict__ cb, float* __restrict__ out) {
    int node = blockIdx.x * 8 + (threadIdx.x >> 5);
    if (node >= T_ * N_) return;
    int lane = threadIdx.x & 31;
    const float4* hp = (const float4*)(h + (size_t)node * D_ + lane * 8);
    float4 h0 = hp[0], h1 = hp[1];
    float x[8] = {h0.x, h0.y, h0.z, h0.w, h1.x, h1.y, h1.z, h1.w};
    float s0 = 0.f, s1 = 0.f;
#pragma unroll
    for (int i = 0; i < 8; ++i) {
        int d = lane * 8 + i;
        s0 += x[i] * cw[d * 2 + 0];
        s1 += x[i] * cw[d * 2 + 1];
    }
#pragma unroll
    for (int off = 16; off; off >>= 1) {
        s0 += __shfl_xor(s0, off, 32);
        s1 += __shfl_xor(s1, off, 32);
    }
    if (lane == 0) {
        out[(size_t)node * 2 + 0] = s0 + cb[0];
        out[(size_t)node * 2 + 1] = s1 + cb[1];
    }
}

// ---------- host orchestration ----------
static inline int cdiv(int a, int b) { return (a + b - 1) / b; }

extern "C" void kernel_launch(void* const* d_in, const int* in_sizes, int n_in,
                              void* d_out, int out_size, void* d_ws, size_t ws_size,
                              hipStream_t stream) {
    (void)in_sizes; (void)n_in; (void)out_size; (void)ws_size;
    const float* feat     = (const float*)d_in[0];
    const int*   src_idx  = (const int*)  d_in[1];
    const int*   dst_idx  = (const int*)  d_in[2];
    const float* adapt_W  = (const float*)d_in[3];
    const float* adapt_b  = (const float*)d_in[4];
    const float* Wk       = (const float*)d_in[5];
    const float* bk       = (const float*)d_in[6];
    const float* Wq       = (const float*)d_in[7];
    const float* bq       = (const float*)d_in[8];
    const float* Wv       = (const float*)d_in[9];
    const float* bv       = (const float*)d_in[10];
    const float* Wa       = (const float*)d_in[11];
    const float* ba       = (const float*)d_in[12];
    const float* rel_att  = (const float*)d_in[13];
    const float* rel_msg  = (const float*)d_in[14];
    const float* rel_pri  = (const float*)d_in[15];
    const float* skipw    = (const float*)d_in[16];
    const float* ln_g     = (const float*)d_in[17];
    const float* ln_b     = (const float*)d_in[18];
    const float* cls_W    = (const float*)d_in[19];
    const float* cls_b    = (const float*)d_in[20];
    float* out = (float*)d_out;

    // workspace layout (floats)
    float* h    = (float*)d_ws;
    float* kb   = h   + TND_;
    float* qb   = kb  + TND_;
    float* vb   = qb  + TND_;
    float* kr   = vb  + TND_;
    float* vr   = kr  + ND_;
    float* agg  = vr  + ND_;
    float* ex   = agg + TND_;
    unsigned* mb = (unsigned*)(ex + EH_);
    float* den  = (float*)(mb + NH_);
    float* trans = kb;  // K buffer is dead after the relation loop; reuse

    static const int REL_SRC[R_] = {0, 1, 0, 1};
    static const int REL_DST[R_] = {1, 0, 0, 1};

    const int MB = cdiv(N_, 32);                 // 1563 M-tiles
    const dim3 ggrid(MB, D_ / 64);               // dense GEMM grid
    const dim3 rgrid(MB, H_);                    // relation GEMM grid

    adapt_kernel<<<cdiv(TND_, 256), 256, 0, stream>>>(feat, adapt_W, adapt_b, h);

    for (int l = 0; l < L_; ++l) {
        for (int t = 0; t < T_; ++t) {
            const size_t wo = (size_t)(l * T_ + t) * D_ * D_;
            const size_t bo = (size_t)(l * T_ + t) * D_;
            gemm_bias_wmma<<<ggrid, 256, 0, stream>>>(h + (size_t)t * ND_, Wk + wo, bk + bo,
                                                      kb + (size_t)t * ND_, N_, 1.0f);
            gemm_bias_wmma<<<ggrid, 256, 0, stream>>>(h + (size_t)t * ND_, Wq + wo, bq + bo,
                                                      qb + (size_t)t * ND_, N_, 1.0f);
            gemm_bias_wmma<<<ggrid, 256, 0, stream>>>(h + (size_t)t * ND_, Wv + wo, bv + bo,
                                                      vb + (size_t)t * ND_, N_, 1.0f);
        }
        fill_f32_kernel<<<cdiv(TND_, 256), 256, 0, stream>>>(agg, 0.0f, TND_);

        for (int r = 0; r < R_; ++r) {
            const int s = REL_SRC[r], d = REL_DST[r];
            const size_t ro = (size_t)(l * R_ + r) * H_ * DK_ * DK_;
            rel_gemm_wmma<<<rgrid, 128, 0, stream>>>(kb + (size_t)s * ND_, rel_att + ro, kr, N_);
            rel_gemm_wmma<<<rgrid, 128, 0, stream>>>(vb + (size_t)s * ND_, rel_msg + ro, vr, N_);
            fill_u32_kernel<<<cdiv(NH_, 256), 256, 0, stream>>>(mb, 0u, NH_);
            fill_f32_kernel<<<cdiv(NH_, 256), 256, 0, stream>>>(den, 0.0f, NH_);
            score_kernel<<<cdiv(E_, 8), 256, 0, stream>>>(
                qb + (size_t)d * ND_, kr, src_idx + (size_t)r * E_, dst_idx + (size_t)r * E_,
                rel_pri + (size_t)(l * R_ + r) * H_, ex, mb, E_);
            exp_den_kernel<<<cdiv(EH_, 256), 256, 0, stream>>>(
                ex, mb, den, dst_idx + (size_t)r * E_, E_);
            aggregate_kernel<<<cdiv(E_, 8), 256, 0, stream>>>(
                ex, den, vr, src_idx + (size_t)r * E_, dst_idx + (size_t)r * E_,
                agg + (size_t)d * ND_, E_);
        }
        for (int t = 0; t < T_; ++t) {
            const size_t wo = (size_t)(l * T_ + t) * D_ * D_;
            const size_t bo = (size_t)(l * T_ + t) * D_;
            // cross_reducer mean (/2) folded into epilogue xscale
            gemm_bias_wmma<<<ggrid, 256, 0, stream>>>(agg + (size_t)t * ND_, Wa + wo, ba + bo,
                                                      trans + (size_t)t * ND_, N_, 0.5f);
        }
        skip_ln_kernel<<<cdiv(T_ * N_, 8), 256, 0, stream>>>(
            trans, h, skipw + (size_t)l * T_,
            ln_g + (size_t)l * T_ * D_, ln_b + (size_t)l * T_ * D_);
    }
    classifier_kernel<<<cdiv(T_ * N_, 8), 256, 0, stream>>>(h, cls_W, cls_b, out);
}